// LQActiv_72928544686741
// MI455X (gfx1250) — compile-verified
//
#include <hip/hip_runtime.h>
#include <stdint.h>

typedef float v4f __attribute__((ext_vector_type(4)));

#define BLOCK 256
#define MAXGRID 4096
#define DEPTH 4   // async pipeline stages (power of 2)

struct QP {
  float t0, t1, t2;
  float e0[4], e1[4];
  float v0, v1;
};

__device__ __forceinline__ QP make_qp(float v0, float v1) {
  float q[4], a[4], b[4];
#pragma unroll
  for (int k = 0; k < 4; ++k) {
    float ea = (k & 2) ? 1.0f : -1.0f;  // ENCODINGS[k][0] = 2*((k>>1)&1)-1
    float eb = (k & 1) ? 1.0f : -1.0f;  // ENCODINGS[k][1] = 2*(k&1)-1
    a[k] = ea; b[k] = eb;
    q[k] = ea * v0 + eb * v1;
  }
  // sort 4 (qlevel, e0, e1) triples ascending by qlevel (5-comparator network).
  // basis is wave-uniform -> compiler lowers this entirely to SALU s_cselect.
#define CSWAP(i, j)                                                      \
  {                                                                      \
    bool c = q[i] > q[j];                                                \
    float tq = c ? q[j] : q[i], ta = c ? a[j] : a[i], tb = c ? b[j] : b[i]; \
    q[j] = c ? q[i] : q[j]; a[j] = c ? a[i] : a[j]; b[j] = c ? b[i] : b[j]; \
    q[i] = tq; a[i] = ta; b[i] = tb;                                     \
  }
  CSWAP(0, 1) CSWAP(2, 3) CSWAP(0, 2) CSWAP(1, 3) CSWAP(1, 2)
#undef CSWAP
  QP p;
  p.t0 = 0.5f * (q[0] + q[1]);
  p.t1 = 0.5f * (q[1] + q[2]);
  p.t2 = 0.5f * (q[2] + q[3]);
#pragma unroll
  for (int k = 0; k < 4; ++k) { p.e0[k] = a[k]; p.e1[k] = b[k]; }
  p.v0 = v0; p.v1 = v1;
  return p;
}

// quantize one value, accumulate the three normal-equation sums
__device__ __forceinline__ float qstep(float w, const QP& p,
                                       float& s0, float& s1, float& sp) {
  bool b0 = w > p.t0, b1 = w > p.t1, b2 = w > p.t2;  // idx = b0+b1+b2 (searchsorted 'left')
  float e0 = p.e0[0], e1 = p.e1[0];
  e0 = b0 ? p.e0[1] : e0;  e1 = b0 ? p.e1[1] : e1;
  e0 = b1 ? p.e0[2] : e0;  e1 = b1 ? p.e1[2] : e1;
  e0 = b2 ? p.e0[3] : e0;  e1 = b2 ? p.e1[3] : e1;
  s0 = fmaf(e0, w, s0);
  s1 = fmaf(e1, w, s1);
  sp += e0 * e1;
  return fmaf(e0, p.v0, e1 * p.v1);
}

// ---- CDNA5 async-to-LDS streaming path ----
__device__ __forceinline__ void async_b128(unsigned lds_off, unsigned goff,
                                           const void* base) {
  asm volatile("global_load_async_to_lds_b128 %0, %1, %2 th:TH_LOAD_NT"
               :: "v"(lds_off), "v"(goff), "s"(base)
               : "memory");
}
__device__ __forceinline__ void wait_async_le(int k) {
  switch (k) {
    case 0:  asm volatile("s_wait_asynccnt 0x0" ::: "memory"); break;
    case 1:  asm volatile("s_wait_asynccnt 0x1" ::: "memory"); break;
    case 2:  asm volatile("s_wait_asynccnt 0x2" ::: "memory"); break;
    default: asm volatile("s_wait_asynccnt 0x3" ::: "memory"); break;
  }
}

__global__ __launch_bounds__(BLOCK)
void lq_main(const float* __restrict__ x, const float* __restrict__ basis,
             float* __restrict__ out, float* __restrict__ wsf,
             int* __restrict__ wsi, long long n) {
  __shared__ v4f sbuf[DEPTH][BLOCK];      // per-thread 4-deep pipeline, 16 KB
  __shared__ float rbuf[3][BLOCK / 32];

  const int tid = threadIdx.x;
  const long long n4 = n >> 2;
  const long long stride = (long long)gridDim.x * BLOCK;
  const long long gid = (long long)blockIdx.x * BLOCK + tid;

  QP p = make_qp(basis[0], basis[1]);
  float s0 = 0.f, s1 = 0.f, sp = 0.f;

  unsigned lds_off[DEPTH];
#pragma unroll
  for (int k = 0; k < DEPTH; ++k)
    lds_off[k] = (unsigned)(uintptr_t)&sbuf[k][tid];  // low 32 bits = LDS byte offset

  // preamble: issue up to DEPTH-1 chunks ahead
  int inflight = 0;
#pragma unroll
  for (int k = 0; k < DEPTH - 1; ++k) {
    long long j = gid + (long long)k * stride;
    if (j < n4) { async_b128(lds_off[k], (unsigned)(j * 16), (const void*)x); ++inflight; }
  }

  long long i = gid;
  int ph = 0;
  while (i < n4) {
    long long pf = i + (long long)(DEPTH - 1) * stride;
    if (pf < n4) {
      async_b128(lds_off[(ph + DEPTH - 1) & (DEPTH - 1)],
                 (unsigned)(pf * 16), (const void*)x);
      ++inflight;
    }
    wait_async_le(inflight - 1);          // loads retire in order -> oldest done
    --inflight;

    v4f w = sbuf[ph][tid];                // ds_load_b128, own slot -> no barrier
    v4f qv;
    qv.x = qstep(w.x, p, s0, s1, sp);
    qv.y = qstep(w.y, p, s0, s1, sp);
    qv.z = qstep(w.z, p, s0, s1, sp);
    qv.w = qstep(w.w, p, s0, s1, sp);
    __builtin_nontemporal_store(qv, (v4f*)out + i);  // NT: don't thrash L2

    i += stride;
    ph = (ph + 1) & (DEPTH - 1);
  }

  // scalar tail (n % 4)
  const long long tail_base = n4 << 2;
  if (gid < (n - tail_base)) {
    float w = x[tail_base + gid];
    out[tail_base + gid] = qstep(w, p, s0, s1, sp);
  }

  // wave32 reduce
#pragma unroll
  for (int off = 16; off > 0; off >>= 1) {
    s0 += __shfl_xor(s0, off, 32);
    s1 += __shfl_xor(s1, off, 32);
    sp += __shfl_xor(sp, off, 32);
  }
  const int wid = tid >> 5, lane = tid & 31;
  if (lane == 0) { rbuf[0][wid] = s0; rbuf[1][wid] = s1; rbuf[2][wid] = sp; }
  __syncthreads();
  if (tid == 0) {
    float a = 0.f, b = 0.f, c = 0.f;
#pragma unroll
    for (int k = 0; k < BLOCK / 32; ++k) {
      a += rbuf[0][k]; b += rbuf[1][k]; c += rbuf[2][k];
    }
    atomicAdd(wsf + 0, a);
    atomicAdd(wsf + 1, b);
    atomicAdd(wsi + 2, (int)c);           // exact signed count
  }
}

__global__ void lq_solve(const float* __restrict__ basis,
                         const float* __restrict__ wsf,
                         const int* __restrict__ wsi,
                         float* __restrict__ out, long long n) {
  if (threadIdx.x == 0 && blockIdx.x == 0) {
    double Sb0 = (double)wsf[0];
    double Sb1 = (double)wsf[1];
    double S01 = (double)wsi[2];
    double N = (double)n;
    double det = N * N - S01 * S01;
    double v0 = (N * Sb0 - S01 * Sb1) / det;
    double v1 = (N * Sb1 - S01 * Sb0) / det;
    out[n]     = (float)(0.9 * (double)basis[0] + 0.1 * v0);
    out[n + 1] = (float)(0.9 * (double)basis[1] + 0.1 * v1);
  }
}

extern "C" void kernel_launch(void* const* d_in, const int* in_sizes, int n_in,
                              void* d_out, int out_size, void* d_ws, size_t ws_size,
                              hipStream_t stream) {
  (void)n_in; (void)out_size; (void)ws_size;
  const float* x     = (const float*)d_in[0];
  const float* basis = (const float*)d_in[1];
  float* out = (float*)d_out;
  float* wsf = (float*)d_ws;
  int*   wsi = (int*)d_ws;
  long long n = (long long)in_sizes[0];

  hipMemsetAsync(d_ws, 0, 3 * sizeof(float), stream);  // capture-safe memset node

  long long n4 = n >> 2;
  long long blocks = (n4 + BLOCK - 1) / BLOCK;
  if (blocks < 1) blocks = 1;
  if (blocks > MAXGRID) blocks = MAXGRID;

  lq_main<<<dim3((unsigned)blocks), dim3(BLOCK), 0, stream>>>(x, basis, out, wsf, wsi, n);
  lq_solve<<<1, 1, 0, stream>>>(basis, wsf, wsi, out, n);
}